// HaarWaveletTransform3D_6502580486721
// MI455X (gfx1250) — compile-verified
//
#include <hip/hip_runtime.h>

// 3D causal Haar wavelet transform, MI455X (gfx1250).
// Bandwidth-bound: 0.42 GB moved, ~0.2 GFLOP -> HBM roofline ~18us @ 23.3 TB/s.
// Data path: global_load_async_to_lds_b128 (ASYNCcnt) -> LDS -> ds_load_b64 ->
// 24-add butterfly -> coalesced b32 stores.

#define HAAR_SCALE 0.3536f

__global__ __launch_bounds__(256)
void haar3d_gfx1250_kernel(const float* __restrict__ x, float* __restrict__ out) {
    constexpr int C  = 3;
    constexpr int T  = 33;
    constexpr int H  = 512;
    constexpr int W  = 512;   // input W (floats)
    constexpr int Tp = 17;
    constexpr int Hp = 256;
    constexpr int Wp = 256;

    // 4 input rows of 512 floats = 8 KB tile
    __shared__ float tile[4 * W];

    const int tid = threadIdx.x;           // 0..255 == output k
    unsigned bid  = blockIdx.x;            // over B*C*Tp*Hp = 26112
    const int j   = bid & (Hp - 1);  bid >>= 8;
    const int i   = bid % Tp;        bid /= Tp;
    const int c   = bid % C;
    const int b   = bid / C;

    // Causal padding: xp = concat(x[:, :, :1], x). Block i uses xp[2i], xp[2i+1]
    // -> input frames t0 = max(2i-1, 0), t1 = 2i.
    const int t1 = 2 * i;
    const int t0 = (i == 0) ? 0 : (2 * i - 1);
    const int h0 = 2 * j;

    const long long bcT = (long long)(b * C + c) * T;

    // LDS byte address = low 32 bits of flat address (aperture rule: addr[31:0]).
    const unsigned lds_base = (unsigned)(unsigned long long)(&tile[0]);

    // 512 x B128 transfers cover the 8KB tile; 2 per lane.
#pragma unroll
    for (int n = 0; n < 2; ++n) {
        const int e       = tid + n * 256;     // 0..511
        const int byteoff = e << 4;            // 16B granules
        const int r       = byteoff >> 11;     // row 0..3 (2048B rows)
        const int rowbyte = byteoff & 2047;
        const int t       = (r & 2) ? t1 : t0;
        const int h       = h0 + (r & 1);
        const long long rowbase = ((bcT + t) * H + h) * (long long)W; // floats
        const unsigned long long gaddr =
            (unsigned long long)(x + rowbase) + (unsigned long long)rowbyte;
        const unsigned lds_addr = lds_base + (unsigned)byteoff;
        asm volatile("global_load_async_to_lds_b128 %0, %1, off"
                     :
                     : "v"(lds_addr), "v"(gaddr)
                     : "memory");
    }

    // Wait for this wave's async transfers, then sync so every wave sees the
    // whole 8KB tile.
    asm volatile("s_wait_asynccnt 0x0" ::: "memory");
    __syncthreads();

    // Each lane owns output column k = tid: needs input w-pair (2k, 2k+1) from
    // all 4 rows. ds_load_b64, bank-conflict-free.
    const float2* t2 = (const float2*)tile;
    const float2 v00 = t2[0 * Wp + tid];   // (t0, h0)  p=0,q=0
    const float2 v01 = t2[1 * Wp + tid];   // (t0, h1)  p=0,q=1
    const float2 v10 = t2[2 * Wp + tid];   // (t1, h0)  p=1,q=0
    const float2 v11 = t2[3 * Wp + tid];   // (t1, h1)  p=1,q=1

    // Stage 1: W butterfly
    const float s00 = v00.x + v00.y, d00 = v00.x - v00.y;
    const float s01 = v01.x + v01.y, d01 = v01.x - v01.y;
    const float s10 = v10.x + v10.y, d10 = v10.x - v10.y;
    const float s11 = v11.x + v11.y, d11 = v11.x - v11.y;

    // Stage 2: H butterfly
    const float hs0 = s00 + s01, hs1 = s00 - s01;   // p=0, sw=0, sh=0/1
    const float hs2 = s10 + s11, hs3 = s10 - s11;   // p=1, sw=0
    const float hd0 = d00 + d01, hd1 = d00 - d01;   // p=0, sw=1
    const float hd2 = d10 + d11, hd3 = d10 - d11;   // p=1, sw=1

    // Stage 3: T butterfly; subband order s = st*4 + sh*2 + sw (lll..hhh)
    float o[8];
    o[0] = hs0 + hs2;   // lll
    o[1] = hd0 + hd2;   // llh
    o[2] = hs1 + hs3;   // lhl
    o[3] = hd1 + hd3;   // lhh
    o[4] = hs0 - hs2;   // hll
    o[5] = hd0 - hd2;   // hlh
    o[6] = hs1 - hs3;   // hhl
    o[7] = hd1 - hd3;   // hhh

    // out[b, s*C + c, i, j, k]; channel stride over s is C*Tp*Hp*Wp.
    const long long base =
        ((((long long)b * 8 * C + c) * Tp + i) << 16) + ((long long)j << 8) + tid;
    constexpr long long sstride = (long long)C * Tp * Hp * Wp;  // 3*17*65536
#pragma unroll
    for (int s = 0; s < 8; ++s) {
        out[base + s * sstride] = o[s] * HAAR_SCALE;
    }
}

extern "C" void kernel_launch(void* const* d_in, const int* in_sizes, int n_in,
                              void* d_out, int out_size, void* d_ws, size_t ws_size,
                              hipStream_t stream) {
    const float* x = (const float*)d_in[0];
    float* out     = (float*)d_out;
    const int grid = 2 * 3 * 17 * 256;  // B*C*Tp*Hp = 26112 blocks
    haar3d_gfx1250_kernel<<<grid, 256, 0, stream>>>(x, out);
}